// GCN_65910568124468
// MI455X (gfx1250) — compile-verified
//
#include <hip/hip_runtime.h>

typedef float v2f __attribute__((ext_vector_type(2)));
typedef float v8f __attribute__((ext_vector_type(8)));

#define CIN 128
#define COUT 64

// ---------------------------------------------------------------- degree prep
__global__ __launch_bounds__(256) void gcn_init_deg(float* deg, int n) {
  int i = blockIdx.x * 256 + threadIdx.x;
  if (i < n) deg[i] = 1.0f;  // self-loop
}

__global__ __launch_bounds__(256) void gcn_count_deg(const long long* __restrict__ dst,
                                                     float* deg, int n_edges) {
  int e = blockIdx.x * 256 + threadIdx.x;
  if (e < n_edges) atomicAdd(&deg[(size_t)dst[e]], 1.0f);
}

__global__ __launch_bounds__(256) void gcn_dinv(const float* __restrict__ deg,
                                                float* __restrict__ dinv, int n) {
  int i = blockIdx.x * 256 + threadIdx.x;
  if (i < n) dinv[i] = rsqrtf(deg[i]);
}

// ---------------------------------------------------------------- H = X @ W2
// Block: 256 threads = 8 waves, 64 rows x 128 cols of H.
// Wave (r,c): rows r*16..r*16+15, cols c*64..c*64+63 (4 accumulators).
__global__ __launch_bounds__(256) void gcn_gemm_xw(const float* __restrict__ X,
                                                   const float* __restrict__ W,
                                                   float* __restrict__ H, int n) {
  __shared__ float Xt[CIN * 65];   // X tile transposed [k][row], pad 65 -> no bank conflicts
  __shared__ float Wl[32 * CIN];   // W K-chunk (32 rows x 128)
  const int tid  = threadIdx.x;
  const int row0 = blockIdx.x * 64;

  // Load + transpose X tile (64 rows x 128) into LDS, rows clamped.
  #pragma unroll
  for (int i = 0; i < 8; ++i) {
    int idx = tid + 256 * i;          // 2048 float4 total
    int r   = idx >> 5;               // 0..63
    int c4  = idx & 31;               // 0..31
    int gr  = row0 + r; if (gr >= n) gr = n - 1;
    float4 v = ((const float4*)(X + (size_t)gr * CIN))[c4];
    int c = c4 * 4;
    Xt[(c + 0) * 65 + r] = v.x;
    Xt[(c + 1) * 65 + r] = v.y;
    Xt[(c + 2) * 65 + r] = v.z;
    Xt[(c + 3) * 65 + r] = v.w;
  }

  const int lane = tid & 31;
  const int wave = tid >> 5;
  const int wrow = (wave >> 1) * 16;
  const int wcol = (wave & 1) * 64;
  const int m    = lane & 15;
  const int koff = (lane >> 4) * 2;   // lanes 0-15: K+0/K+1, lanes 16-31: K+2/K+3

  v8f c0 = {}, c1 = {}, c2 = {}, c3 = {};

  for (int kc = 0; kc < CIN; kc += 32) {
    __syncthreads();
    {  // stage W rows [kc, kc+32)
      const float4* Wg4 = (const float4*)W;
      float4* Wl4 = (float4*)Wl;
      #pragma unroll
      for (int i = 0; i < 4; ++i) {
        int idx = tid + 256 * i;      // 1024 float4
        Wl4[idx] = Wg4[kc * (CIN / 4) + idx];
      }
    }
    __syncthreads();
    #pragma unroll
    for (int k0 = 0; k0 < 32; k0 += 4) {
      int kk = kc + k0 + koff;
      v2f a;
      a.x = Xt[kk * 65 + wrow + m];
      a.y = Xt[(kk + 1) * 65 + wrow + m];
      const float* wr0 = &Wl[(k0 + koff) * CIN + wcol + m];
      const float* wr1 = wr0 + CIN;
      v2f b;
      b.x = wr0[0];  b.y = wr1[0];
      c0 = __builtin_amdgcn_wmma_f32_16x16x4_f32(false, a, false, b, (short)0, c0, false, false);
      b.x = wr0[16]; b.y = wr1[16];
      c1 = __builtin_amdgcn_wmma_f32_16x16x4_f32(false, a, false, b, (short)0, c1, false, false);
      b.x = wr0[32]; b.y = wr1[32];
      c2 = __builtin_amdgcn_wmma_f32_16x16x4_f32(false, a, false, b, (short)0, c2, false, false);
      b.x = wr0[48]; b.y = wr1[48];
      c3 = __builtin_amdgcn_wmma_f32_16x16x4_f32(false, a, false, b, (short)0, c3, false, false);
    }
  }

  // C/D layout: lanes 0-15 -> M=r, lanes 16-31 -> M=8+r, N = lane&15
  const int rbase = row0 + wrow + (lane >> 4) * 8;
  #pragma unroll
  for (int r = 0; r < 8; ++r) {
    int gr = rbase + r;
    if (gr < n) {
      float* h = H + (size_t)gr * CIN + wcol + m;
      h[0] = c0[r]; h[16] = c1[r]; h[32] = c2[r]; h[48] = c3[r];
    }
  }
}

// ------------------------------------------------- OUT = dinv^2 * H (self loop)
__global__ __launch_bounds__(256) void gcn_selfloop(const float* __restrict__ H,
                                                    const float* __restrict__ dinv,
                                                    float* __restrict__ OUT, int n) {
  int idx = blockIdx.x * 256 + threadIdx.x;   // float4 index, n*32 total
  int v = idx >> 5;
  if (v < n) {
    float w = dinv[v] * dinv[v];
    float4 h = ((const float4*)H)[idx];
    float4 o; o.x = h.x * w; o.y = h.y * w; o.z = h.z * w; o.w = h.w * w;
    ((float4*)OUT)[idx] = o;
  }
}

// --------------------------------------------- edge scatter: OUT[d] += w*H[s]
__global__ __launch_bounds__(256) void gcn_scatter(const long long* __restrict__ src,
                                                   const long long* __restrict__ dst,
                                                   const float* __restrict__ dinv,
                                                   const float* __restrict__ H,
                                                   float* __restrict__ OUT, int n_edges) {
  int e = blockIdx.x * 8 + (threadIdx.x >> 5);   // one wave per edge
  if (e >= n_edges) return;
  int lane = threadIdx.x & 31;
  long long s = src[e];
  long long d = dst[e];
  float w = dinv[s] * dinv[d];
  float4 h = ((const float4*)(H + (size_t)s * CIN))[lane];
  float* o = OUT + (size_t)d * CIN + lane * 4;
  atomicAdd(o + 0, h.x * w);
  atomicAdd(o + 1, h.y * w);
  atomicAdd(o + 2, h.z * w);
  atomicAdd(o + 3, h.w * w);
}

// ------------------------- out = softmax((AGG+b2) @ Wout + bout), fused WMMA
// Block: 256 threads = 8 waves, each wave: 16 rows x all 64 cols.
__global__ __launch_bounds__(256) void gcn_out_softmax(const float* __restrict__ AGG,
                                                       const float* __restrict__ b2,
                                                       const float* __restrict__ Wout,
                                                       const float* __restrict__ bout,
                                                       float* __restrict__ out, int n) {
  __shared__ float Wl[CIN * COUT];   // 32 KB
  __shared__ float b2l[CIN];
  __shared__ float boutl[COUT];
  const int tid  = threadIdx.x;
  const int row0 = blockIdx.x * 128;
  {
    const float4* Wg4 = (const float4*)Wout;
    float4* Wl4 = (float4*)Wl;
    #pragma unroll
    for (int i = 0; i < 8; ++i) Wl4[tid + 256 * i] = Wg4[tid + 256 * i];
  }
  if (tid < CIN)  b2l[tid]  = b2[tid];
  if (tid < COUT) boutl[tid] = bout[tid];
  __syncthreads();

  const int lane = tid & 31;
  const int wave = tid >> 5;
  const int wrow = wave * 16;
  const int m    = lane & 15;
  const int koff = (lane >> 4) * 2;

  int ar = row0 + wrow + m; if (ar >= n) ar = n - 1;   // clamped A row
  const float* arow = AGG + (size_t)ar * CIN;

  v8f c0 = {}, c1 = {}, c2 = {}, c3 = {};
  #pragma unroll 8
  for (int k0 = 0; k0 < CIN; k0 += 4) {
    int kk = k0 + koff;
    v2f a;
    a.x = arow[kk]     + b2l[kk];
    a.y = arow[kk + 1] + b2l[kk + 1];
    const float* w0 = &Wl[kk * COUT + m];
    const float* w1 = w0 + COUT;
    v2f b;
    b.x = w0[0];  b.y = w1[0];
    c0 = __builtin_amdgcn_wmma_f32_16x16x4_f32(false, a, false, b, (short)0, c0, false, false);
    b.x = w0[16]; b.y = w1[16];
    c1 = __builtin_amdgcn_wmma_f32_16x16x4_f32(false, a, false, b, (short)0, c1, false, false);
    b.x = w0[32]; b.y = w1[32];
    c2 = __builtin_amdgcn_wmma_f32_16x16x4_f32(false, a, false, b, (short)0, c2, false, false);
    b.x = w0[48]; b.y = w1[48];
    c3 = __builtin_amdgcn_wmma_f32_16x16x4_f32(false, a, false, b, (short)0, c3, false, false);
  }

  // Softmax: row M=r lives on lanes 0-15, row M=8+r on lanes 16-31.
  // shfl_xor masks 1..8 never cross the 16-lane halves on wave32.
  const int rbase = row0 + wrow + (lane >> 4) * 8;
  #pragma unroll
  for (int r = 0; r < 8; ++r) {
    float v0 = c0[r] + boutl[m];
    float v1 = c1[r] + boutl[16 + m];
    float v2 = c2[r] + boutl[32 + m];
    float v3 = c3[r] + boutl[48 + m];
    float mx = fmaxf(fmaxf(v0, v1), fmaxf(v2, v3));
    #pragma unroll
    for (int msk = 1; msk < 16; msk <<= 1)
      mx = fmaxf(mx, __shfl_xor(mx, msk, 32));
    float e0 = __expf(v0 - mx), e1 = __expf(v1 - mx);
    float e2 = __expf(v2 - mx), e3 = __expf(v3 - mx);
    float s = e0 + e1 + e2 + e3;
    #pragma unroll
    for (int msk = 1; msk < 16; msk <<= 1)
      s += __shfl_xor(s, msk, 32);
    int gr = rbase + r;
    if (gr < n) {
      float inv = 1.0f / s;
      float* o = out + (size_t)gr * COUT + m;
      o[0] = e0 * inv; o[16] = e1 * inv; o[32] = e2 * inv; o[48] = e3 * inv;
    }
  }
}

// ---------------------------------------------------------------------- host
extern "C" void kernel_launch(void* const* d_in, const int* in_sizes, int n_in,
                              void* d_out, int out_size, void* d_ws, size_t ws_size,
                              hipStream_t stream) {
  const float*     x    = (const float*)d_in[0];
  const long long* ei   = (const long long*)d_in[1];
  const float*     W2   = (const float*)d_in[5];   // i == 1 path
  const float*     b2   = (const float*)d_in[6];
  const float*     Wout = (const float*)d_in[7];
  const float*     bout = (const float*)d_in[8];
  float* out = (float*)d_out;

  const int N = in_sizes[0] / CIN;
  const int E = in_sizes[1] / 2;
  const long long* srcp = ei;
  const long long* dstp = ei + E;

  float* H    = (float*)d_ws;                 // N*128
  float* AGG  = H   + (size_t)N * CIN;        // N*128
  float* deg  = AGG + (size_t)N * CIN;        // N
  float* dinv = deg + N;                      // N

  gcn_init_deg   <<<(N + 255) / 256, 256, 0, stream>>>(deg, N);
  gcn_count_deg  <<<(E + 255) / 256, 256, 0, stream>>>(dstp, deg, E);
  gcn_dinv       <<<(N + 255) / 256, 256, 0, stream>>>(deg, dinv, N);
  gcn_gemm_xw    <<<(N + 63) / 64,   256, 0, stream>>>(x, W2, H, N);
  gcn_selfloop   <<<(N * 32 + 255) / 256, 256, 0, stream>>>(H, dinv, AGG, N);
  gcn_scatter    <<<(E + 7) / 8,     256, 0, stream>>>(srcp, dstp, dinv, H, AGG, E);
  gcn_out_softmax<<<(N + 127) / 128, 256, 0, stream>>>(AGG, b2, Wout, bout, out, N);

  (void)n_in; (void)out_size; (void)ws_size;
}